// BahdanauAttention_77953656422604
// MI455X (gfx1250) — compile-verified
//
#include <hip/hip_runtime.h>
#include <hip/hip_bf16.h>
#include <math.h>

// Problem constants (fixed by the reference): B=64, T=4096, QD=1024, AD=256
#define BB 64
#define TT 4096
#define QD 1024
#define AD 256
#define SCORE_MASK_VALUE (-1e30f)

typedef __attribute__((ext_vector_type(2))) float v2f;
typedef __attribute__((ext_vector_type(4))) float f4;
typedef __attribute__((ext_vector_type(8))) float v8f;

#if __has_builtin(__builtin_amdgcn_tanhf)
#define TANHF(x) __builtin_amdgcn_tanhf(x)
#else
#define TANHF(x) tanhf(x)
#endif

// ---------------------------------------------------------------------------
// Kernel 1: processed_query = query @ Wq^T  via V_WMMA_F32_16X16X4_F32.
// One wave (32 threads) per 16x16 output tile. grid = (B/16, AD/16) = (4,16).
//
// A (16x4 f32, M x K) lane layout (ISA 7.12.2): lanes 0-15 hold K={k,k+1},
// lanes 16-31 hold K={k+2,k+3}; row M = lane&15.  B (4x16, K x N) mirrors it:
// col N = lane&15, same K split.  C/D: VGPR r -> row r + (lane>=16)*8,
// col = lane&15.
// ---------------------------------------------------------------------------
__global__ __launch_bounds__(32) void pq_wmma_kernel(
    const float* __restrict__ q,    // (B, QD) row-major
    const float* __restrict__ Wq,   // (AD, QD) row-major
    float* __restrict__ pq)         // (B, AD) row-major
{
    const int m0   = blockIdx.x * 16;          // batch tile
    const int n0   = blockIdx.y * 16;          // AD tile
    const int lane = threadIdx.x;              // 0..31
    const int l15  = lane & 15;
    const int half = lane >> 4;                // 0 or 1

    // A row for this lane: query[m0+l15][*]; B "column" d = n0+l15: Wq[d][*]
    const v2f* qa = (const v2f*)(q  + (size_t)(m0 + l15) * QD);
    const v2f* wb = (const v2f*)(Wq + (size_t)(n0 + l15) * QD);

    v8f c = {};
#pragma unroll 4
    for (int k = 0; k < QD; k += 4) {
        const int kh = (k >> 1) + half;        // float2 index: k + 2*half
        v2f a = qa[kh];
        v2f b = wb[kh];
        // (neg_a, A, neg_b, B, c_mod, C, reuse_a, reuse_b)
        c = __builtin_amdgcn_wmma_f32_16x16x4_f32(
                false, a, false, b, (short)0, c, false, false);
    }

#pragma unroll
    for (int r = 0; r < 8; ++r) {
        const int row = m0 + r + half * 8;
        pq[(size_t)row * AD + (n0 + l15)] = c[r];
    }
}

// ---------------------------------------------------------------------------
// Kernel 2: energies[b][t] = mask ? -1e30 : sum_d v[d]*tanh(pq[b][d]+pm[b][t][d])
// Bandwidth-bound streamer over pm (268 MB): non-temporal b128 loads, each
// instruction covers 512 contiguous bytes across the wave. Lane l owns AD
// indices {4l..4l+3, 128+4l..128+4l+3}. grid = (T/128, B), 256 thr (8 waves),
// each wave processes 16 consecutive t rows.
// ---------------------------------------------------------------------------
__global__ __launch_bounds__(256) void energies_kernel(
    const float* __restrict__ pm,            // (B, T, AD)
    const unsigned char* __restrict__ mask,  // (B, T) bool (1 byte/elem)
    const float* __restrict__ pq,            // (B, AD)
    const float* __restrict__ v,             // (AD)
    float* __restrict__ out)                 // (B, T) energies (pre-softmax)
{
    const int b    = blockIdx.y;
    const int lane = threadIdx.x & 31;
    const int wave = threadIdx.x >> 5;

    // Per-lane constant fragments (live in registers for all 16 rows).
    const f4* pq4 = (const f4*)(pq + (size_t)b * AD);
    const f4* v4  = (const f4*)v;
    const f4 q0 = pq4[lane];
    const f4 q1 = pq4[lane + 32];
    const f4 w0 = v4[lane];
    const f4 w1 = v4[lane + 32];

    const int    tbase   = blockIdx.x * 128 + wave * 16;
    const size_t rowbase = ((size_t)b * TT + tbase) * AD;

    for (int i = 0; i < 16; ++i) {
        const int t = tbase + i;
        const f4* row = (const f4*)(pm + rowbase + (size_t)i * AD);
        if (i < 12)  // software prefetch a few rows ahead (global_prefetch_b8)
            __builtin_prefetch(pm + rowbase + (size_t)(i + 4) * AD + 4 * lane, 0, 0);

        // Streaming (read-once) data: non-temporal so it doesn't evict L2.
        const f4 p0 = __builtin_nontemporal_load(row + lane);
        const f4 p1 = __builtin_nontemporal_load(row + lane + 32);

        float e = w0.x * TANHF(q0.x + p0.x);
        e += w0.y * TANHF(q0.y + p0.y);
        e += w0.z * TANHF(q0.z + p0.z);
        e += w0.w * TANHF(q0.w + p0.w);
        e += w1.x * TANHF(q1.x + p1.x);
        e += w1.y * TANHF(q1.y + p1.y);
        e += w1.z * TANHF(q1.z + p1.z);
        e += w1.w * TANHF(q1.w + p1.w);

        // wave32 butterfly reduction
#pragma unroll
        for (int off = 16; off >= 1; off >>= 1)
            e += __shfl_xor(e, off, 32);

        if (lane == 0) {
            const size_t idx = (size_t)b * TT + t;
            out[idx] = mask[idx] ? SCORE_MASK_VALUE : e;
        }
    }
}

// ---------------------------------------------------------------------------
// Kernel 3: row softmax over T=4096, in place on d_out. One block (256 thr)
// per batch row; each thread holds 16 values in registers, so all reads
// complete before any writes (safe in-place).
// ---------------------------------------------------------------------------
__global__ __launch_bounds__(256) void softmax_kernel(float* __restrict__ out)
{
    __shared__ float red[8];
    const int b   = blockIdx.x;
    const int tid = threadIdx.x;
    float* row = out + (size_t)b * TT;

    float vals[16];
    float m = -INFINITY;
#pragma unroll
    for (int i = 0; i < 16; ++i) {
        vals[i] = row[tid + 256 * i];
        m = fmaxf(m, vals[i]);
    }
#pragma unroll
    for (int off = 16; off >= 1; off >>= 1)
        m = fmaxf(m, __shfl_xor(m, off, 32));
    if ((tid & 31) == 0) red[tid >> 5] = m;
    __syncthreads();
    float bm = red[0];
#pragma unroll
    for (int w = 1; w < 8; ++w) bm = fmaxf(bm, red[w]);
    __syncthreads();

    float s = 0.0f;
#pragma unroll
    for (int i = 0; i < 16; ++i) {
        vals[i] = __expf(vals[i] - bm);
        s += vals[i];
    }
#pragma unroll
    for (int off = 16; off >= 1; off >>= 1)
        s += __shfl_xor(s, off, 32);
    if ((tid & 31) == 0) red[tid >> 5] = s;
    __syncthreads();
    float bs = 0.0f;
#pragma unroll
    for (int w = 0; w < 8; ++w) bs += red[w];

    const float inv = 1.0f / bs;
#pragma unroll
    for (int i = 0; i < 16; ++i)
        row[tid + 256 * i] = vals[i] * inv;
}

// ---------------------------------------------------------------------------
// setup_inputs() order: query(f32 B*QD), processed_memory(f32 B*T*AD),
// mask(bool B*T), Wq(f32 AD*QD), v(f32 AD). Output: alignment(f32 B*T).
// ---------------------------------------------------------------------------
extern "C" void kernel_launch(void* const* d_in, const int* in_sizes, int n_in,
                              void* d_out, int out_size, void* d_ws, size_t ws_size,
                              hipStream_t stream) {
    const float*         query = (const float*)d_in[0];
    const float*         pm    = (const float*)d_in[1];
    const unsigned char* mask  = (const unsigned char*)d_in[2]; // jax bool: 1B/elem
    const float*         Wq    = (const float*)d_in[3];
    const float*         v     = (const float*)d_in[4];
    float*               out   = (float*)d_out;
    float*               pqbuf = (float*)d_ws;                  // 64*256 f32 = 64 KB

    // 1) processed_query via fp32 WMMA: grid (B/16, AD/16), one wave per tile.
    pq_wmma_kernel<<<dim3(BB / 16, AD / 16), 32, 0, stream>>>(query, Wq, pqbuf);

    // 2) energies + mask: grid (T/128, B), 8 waves x 16 rows each.
    energies_kernel<<<dim3(TT / 128, BB), 256, 0, stream>>>(pm, mask, pqbuf, v, out);

    // 3) row softmax in place.
    softmax_kernel<<<BB, 256, 0, stream>>>(out);
}